// TrustGNN_80023830659560
// MI455X (gfx1250) — compile-verified
//
#include <hip/hip_runtime.h>

typedef __attribute__((ext_vector_type(2))) float v2f;
typedef __attribute__((ext_vector_type(8))) float v8f;

#define N_NODES 50000
#define N_EDGES 800000
#define IN_DIM  128
#define HID_DIM 256
#define OUT_DIM 128

// ---------------- utility: zero workspace ----------------
__global__ void zero_kernel(float* __restrict__ p, long n) {
    long i = (long)blockIdx.x * blockDim.x + threadIdx.x;
    long stride = (long)gridDim.x * blockDim.x;
    for (; i < n; i += stride) p[i] = 0.0f;
}

// ---------------- degree: deg[dst[e]] += 1 ----------------
__global__ void degree_kernel(const int* __restrict__ dst, float* __restrict__ deg, int nE) {
    int e = blockIdx.x * blockDim.x + threadIdx.x;
    if (e < nE) atomicAdd(&deg[dst[e]], 1.0f);
}

// ---------------- edge scatter: acc[dst[e]] += feat[src[e]] ----------------
// D/4 threads cooperate per edge; each moves one float4 and does 4 f32 atomics.
template<int D>
__global__ __launch_bounds__(256)
void scatter_sum_kernel(const float* __restrict__ feat,
                        const int*   __restrict__ src,
                        const int*   __restrict__ dst,
                        float*       __restrict__ acc, int nE) {
    constexpr int TPE = D / 4;
    long gid = (long)blockIdx.x * blockDim.x + threadIdx.x;
    int e = (int)(gid / TPE);
    int t = (int)(gid % TPE);
    if (e >= nE) return;
    int s = src[e];
    int d = dst[e];
    const float4 v = *reinterpret_cast<const float4*>(feat + (long)s * D + t * 4);
    float* o = acc + (long)d * D + t * 4;
    atomicAdd(o + 0, v.x);
    atomicAdd(o + 1, v.y);
    atomicAdd(o + 2, v.z);
    atomicAdd(o + 3, v.w);
}

// ---------------- fused SAGE layer via V_WMMA_F32_16X16X4_F32 ----------------
// out[m, n] = relu?( sum_k (agg_sum[m,k] * invdeg[m]) * Wl[n,k]
//                  + sum_k self[m,k] * Wr[n,k] + bias[n] )
// One wave32 computes one 16x16 output tile; K loop steps by 4, two WMMAs/step.
//
// fp32 WMMA operand layout (ISA 7.12.2):
//   A 16x4:  lanes 0-15 -> M=lane,  VGPR0=K0, VGPR1=K1; lanes 16-31 -> K2,K3
//   B 4x16:  lanes 0-15 -> N=lane,  VGPR0=K0, VGPR1=K1; lanes 16-31 -> K2,K3
//   C 16x16: VGPR r: lanes 0-15 -> M=r, N=lane; lanes 16-31 -> M=r+8
template<int DIN, int DOUT, bool RELU>
__global__ __launch_bounds__(256)
void sage_wmma_gemm(const float* __restrict__ self_feat,
                    const float* __restrict__ agg_sum,
                    const float* __restrict__ deg,
                    const float* __restrict__ Wl,
                    const float* __restrict__ Wr,
                    const float* __restrict__ bias,
                    float*       __restrict__ out,
                    int nTiles) {
    constexpr int COL_TILES = DOUT / 16;
    const int wave = threadIdx.x >> 5;
    const int tile = blockIdx.x * (blockDim.x >> 5) + wave;
    if (tile >= nTiles) return;          // wave-uniform: EXEC stays all-1s for WMMA

    const int rowTile = tile / COL_TILES;
    const int colTile = tile % COL_TILES;
    const int row0 = rowTile * 16;
    const int col0 = colTile * 16;

    const int lane = threadIdx.x & 31;
    const int half = lane >> 4;          // 0: K={0,1}, 1: K={2,3}
    const int l15  = lane & 15;
    const int koff = half * 2;

    const float* aggp  = agg_sum   + (long)(row0 + l15) * DIN + koff;
    const float* slfp  = self_feat + (long)(row0 + l15) * DIN + koff;
    const float* wlp   = Wl        + (long)(col0 + l15) * DIN + koff;
    const float* wrp   = Wr        + (long)(col0 + l15) * DIN + koff;

    const float dv   = deg[row0 + l15];
    const float invd = 1.0f / fmaxf(dv, 1.0f);   // mean-agg folded into A load

    v8f c = {0.f, 0.f, 0.f, 0.f, 0.f, 0.f, 0.f, 0.f};

#pragma unroll 4
    for (int k = 0; k < DIN; k += 4) {
        v2f a_agg = *reinterpret_cast<const v2f*>(aggp + k);
        v2f a_slf = *reinterpret_cast<const v2f*>(slfp + k);
        v2f b_wl  = *reinterpret_cast<const v2f*>(wlp + k);
        v2f b_wr  = *reinterpret_cast<const v2f*>(wrp + k);
        a_agg *= invd;
        c = __builtin_amdgcn_wmma_f32_16x16x4_f32(false, a_agg, false, b_wl,
                                                  (short)0, c, false, false);
        c = __builtin_amdgcn_wmma_f32_16x16x4_f32(false, a_slf, false, b_wr,
                                                  (short)0, c, false, false);
    }

    const float bcol = bias[col0 + l15];         // column fixed per lane
#pragma unroll
    for (int r = 0; r < 8; ++r) {
        const int row = row0 + r + half * 8;
        float v = c[r] + bcol;
        if (RELU) v = fmaxf(v, 0.0f);
        out[(long)row * DOUT + col0 + l15] = v;
    }
}

// ---------------- launcher ----------------
extern "C" void kernel_launch(void* const* d_in, const int* in_sizes, int n_in,
                              void* d_out, int out_size, void* d_ws, size_t ws_size,
                              hipStream_t stream) {
    const float* x   = (const float*)d_in[0];
    const int*   ei  = (const int*)  d_in[1];   // [2, E] int32 (JAX default int)
    const float* W1l = (const float*)d_in[2];   // [256,128]
    const float* b1  = (const float*)d_in[3];   // [256]
    const float* W1r = (const float*)d_in[4];   // [256,128]
    const float* W2l = (const float*)d_in[5];   // [128,256]
    const float* b2  = (const float*)d_in[6];   // [128]
    const float* W2r = (const float*)d_in[7];   // [128,256]
    float* out = (float*)d_out;                 // [50000,128]

    const int* src = ei;            // row 0: source j
    const int* dst = ei + N_EDGES;  // row 1: target i

    // workspace layout (floats), each region 16B-aligned
    float* ws   = (float*)d_ws;
    float* deg  = ws;                                   // [N]
    float* agg1 = deg  + N_NODES;                       // [N, 128]
    float* agg2 = agg1 + (long)N_NODES * IN_DIM;        // [N, 256]
    float* h    = agg2 + (long)N_NODES * HID_DIM;       // [N, 256]

    // 1) zero deg + agg1 + agg2 (h is fully overwritten)
    const long zcount = (long)N_NODES * (1 + IN_DIM + HID_DIM);
    zero_kernel<<<2048, 256, 0, stream>>>(ws, zcount);

    // 2) degrees (shared by both layers)
    degree_kernel<<<(N_EDGES + 255) / 256, 256, 0, stream>>>(dst, deg, N_EDGES);

    // 3) layer-1 neighbor sum: agg1[dst] += x[src]
    {
        const long threads = (long)N_EDGES * (IN_DIM / 4);
        scatter_sum_kernel<IN_DIM><<<(int)(threads / 256), 256, 0, stream>>>(
            x, src, dst, agg1, N_EDGES);
    }

    // 4) h = relu(mean(agg1) @ W1l^T + b1 + x @ W1r^T)   [fused WMMA]
    {
        const int tiles = (N_NODES / 16) * (HID_DIM / 16);  // 50000, /8 waves exact
        sage_wmma_gemm<IN_DIM, HID_DIM, true><<<tiles / 8, 256, 0, stream>>>(
            x, agg1, deg, W1l, W1r, b1, h, tiles);
    }

    // 5) layer-2 neighbor sum: agg2[dst] += h[src]
    {
        const long threads = (long)N_EDGES * (HID_DIM / 4);
        scatter_sum_kernel<HID_DIM><<<(int)(threads / 256), 256, 0, stream>>>(
            h, src, dst, agg2, N_EDGES);
    }

    // 6) out = mean(agg2) @ W2l^T + b2 + h @ W2r^T   [fused WMMA, no relu]
    {
        const int tiles = (N_NODES / 16) * (OUT_DIM / 16);  // 25000, /8 waves exact
        sage_wmma_gemm<HID_DIM, OUT_DIM, false><<<tiles / 8, 256, 0, stream>>>(
            h, agg2, deg, W2l, W2r, b2, out, tiles);
    }
}